// DoRA_APN_88132728914549
// MI455X (gfx1250) — compile-verified
//
#include <hip/hip_runtime.h>
#include <math.h>

typedef __attribute__((ext_vector_type(2))) float v2f;
typedef __attribute__((ext_vector_type(8))) float v8f;

#define T_    2
#define B_    32
#define D_    768
#define HEADS 12
#define HD    64
#define HT    14
#define WT    14
#define HS    96
#define WS    96
#define NT    (HT*WT)     // 196
#define NS    (HS*WS)     // 9216
#define SM    (HS-HT+1)   // 83

// ---------------------------------------------------------------------------
// Stage 1: masked prototype pooling.  proto[b][d] = mean_t( masked_mean_n(patches) )
// 32 blocks x 256 threads; each thread owns 3 of the 768 channels.
// ---------------------------------------------------------------------------
__global__ void proto_kernel(const float* __restrict__ patches,
                             const float* __restrict__ annos,
                             float* __restrict__ proto) {
    const int b   = blockIdx.x;
    const int tid = threadIdx.x;

    float acc[T_][3];
    float cnt[T_];

    for (int t = 0; t < T_; ++t) {
        const float* a = annos + (size_t)(t * B_ + b) * 4;
        const float a0 = a[0], a1 = a[1], a2 = a[2], a3 = a[3];
        const int x1 = (int)floorf(a0 * (float)WT);
        const int y1 = (int)floorf(a1 * (float)HT);
        const int x2 = (int)floorf((a0 + a2) * (float)WT);
        const int y2 = (int)floorf((a1 + a3) * (float)HT);

        float s0 = 0.f, s1 = 0.f, s2 = 0.f, c = 0.f;
        const float* pb = patches + (size_t)(t * B_ + b) * NT * D_;
        for (int n = 0; n < NT; ++n) {
            const int row = n / WT, col = n % WT;
            const bool m = (row >= y1) && (row < y2) && (col >= x1) && (col < x2);
            if (m) {
                c += 1.0f;
                const float* p = pb + (size_t)n * D_;
                s0 += p[tid];
                s1 += p[tid + 256];
                s2 += p[tid + 512];
            }
        }
        acc[t][0] = s0; acc[t][1] = s1; acc[t][2] = s2; cnt[t] = c;
    }

    const float i0 = 1.0f / (cnt[0] + 1e-6f);
    const float i1 = 1.0f / (cnt[1] + 1e-6f);
    float* pr = proto + (size_t)b * D_;
    pr[tid]       = 0.5f * (acc[0][0] * i0 + acc[1][0] * i1);
    pr[tid + 256] = 0.5f * (acc[0][1] * i0 + acc[1][1] * i1);
    pr[tid + 512] = 0.5f * (acc[0][2] * i0 + acc[1][2] * i1);
}

// ---------------------------------------------------------------------------
// Stage 2: track[b][n] = (1/heads) * hd^-0.5 * sum_h dot(proto_h[b,h], key[b,h,n+1])
// via V_WMMA_F32_16X16X4_F32.  One wave per 16-output tile; the 12-head sum
// accumulates in one f32 16x16 accumulator (A rows replicated -> D rows equal).
// Layouts per CDNA5 ISA 7.12.2:
//   A (16x4 f32): lane L, vgpr v -> (M=L%16, K=v+2*(L>=16))
//   B (4x16 f32): lane L, vgpr v -> (N=L%16, K=v+2*(L>=16))
//   C (16x16 f32): vgpr0 lanes 0-15 -> (M=0, N=lane)
// ---------------------------------------------------------------------------
__global__ void attn_wmma_kernel(const float* __restrict__ skey,
                                 const float* __restrict__ proto,
                                 float* __restrict__ track) {
    __shared__ float s_proto[D_];
    const int b   = blockIdx.y;
    const int tid = threadIdx.x;
    for (int i = tid; i < D_; i += 256) s_proto[i] = proto[(size_t)b * D_ + i];
    __syncthreads();

    const int wave = tid >> 5;
    const int lane = tid & 31;
    const int half = lane >> 4;   // 0: K=v+0, 1: K=v+2
    const int l16  = lane & 15;   // N / row select
    const int nb   = (blockIdx.x * 8 + wave) * 16;

    v8f c = {};
    for (int h = 0; h < HEADS; ++h) {
        const float* krow = skey +
            ((((size_t)b * HEADS + h) * (size_t)(NS + 1)) + 1 + (size_t)(nb + l16)) * HD;
        const float* prot = s_proto + h * HD;
        for (int kb = 0; kb < HD; kb += 4) {
            const int k0 = kb + half * 2;
            v2f av; av.x = prot[k0]; av.y = prot[k0 + 1];   // broadcast from LDS
            v2f bv; bv.x = krow[k0]; bv.y = krow[k0 + 1];   // global_load_b64
            c = __builtin_amdgcn_wmma_f32_16x16x4_f32(
                    false, av, false, bv, (short)0, c, false, false);
        }
    }
    const float scale = 0.125f / 12.0f;   // hd^-0.5 / heads
    if (lane < 16) track[(size_t)b * NS + nb + lane] = c[0] * scale;
}

// ---------------------------------------------------------------------------
// Stage 3: vertical 5-tap Gaussian blur (sigma=1.1), reflect padding.
// ---------------------------------------------------------------------------
__device__ __forceinline__ int reflect_i(int i, int n) {
    if (i < 0)  i = -i;
    if (i >= n) i = 2 * n - 2 - i;
    return i;
}

__global__ void blur_v_kernel(const float* __restrict__ track,
                              float* __restrict__ blurV) {
    const int idx = blockIdx.x * 256 + threadIdx.x;
    if (idx >= B_ * NS) return;
    const int b = idx / NS, r = idx % NS;
    const int y = r / WS, x = r % WS;

    const float e1 = expf(-1.0f / 2.42f);          // 2*sigma^2 = 2.42
    const float e2 = expf(-4.0f / 2.42f);
    const float sn = 1.0f / (1.0f + 2.0f * e1 + 2.0f * e2);
    const float w0 = e2 * sn, w1 = e1 * sn, w2 = sn;

    const float* tb = track + (size_t)b * NS;
    const float v = w0 * tb[reflect_i(y - 2, HS) * WS + x]
                  + w1 * tb[reflect_i(y - 1, HS) * WS + x]
                  + w2 * tb[y * WS + x]
                  + w1 * tb[reflect_i(y + 1, HS) * WS + x]
                  + w0 * tb[reflect_i(y + 2, HS) * WS + x];
    blurV[idx] = v;
}

// ---------------------------------------------------------------------------
// Stage 4 (per-b, LDS resident): horizontal blur -> softmax -> 2D cumsum ->
// 14x14 box-sum argmax.  Outputs written as floats, concatenated in return
// order: best_x[32], best_y[32], win_w[32], win_h[32], confidence[32],
// prob[32*9216].
// ---------------------------------------------------------------------------
__global__ void finalize_kernel(const float* __restrict__ blurV,
                                float* __restrict__ out) {
    __shared__ float img[NS];       // 36 KB
    __shared__ float redv[256];
    __shared__ int   redi[256];

    const int b   = blockIdx.x;
    const int tid = threadIdx.x;

    const float e1 = expf(-1.0f / 2.42f);
    const float e2 = expf(-4.0f / 2.42f);
    const float sn = 1.0f / (1.0f + 2.0f * e1 + 2.0f * e2);
    const float w0 = e2 * sn, w1 = e1 * sn, w2 = sn;

    // horizontal blur into LDS
    const float* vb = blurV + (size_t)b * NS;
    for (int i = tid; i < NS; i += 256) {
        const int y = i / WS, x = i % WS;
        const float* row = vb + y * WS;
        img[i] = w0 * row[reflect_i(x - 2, WS)]
               + w1 * row[reflect_i(x - 1, WS)]
               + w2 * row[x]
               + w1 * row[reflect_i(x + 1, WS)]
               + w0 * row[reflect_i(x + 2, WS)];
    }
    __syncthreads();

    // softmax: max
    float m = -INFINITY;
    for (int i = tid; i < NS; i += 256) m = fmaxf(m, img[i]);
    redv[tid] = m; __syncthreads();
    for (int s = 128; s > 0; s >>= 1) {
        if (tid < s) redv[tid] = fmaxf(redv[tid], redv[tid + s]);
        __syncthreads();
    }
    const float M = redv[0];
    __syncthreads();

    // softmax: exp + sum
    float sum = 0.f;
    for (int i = tid; i < NS; i += 256) {
        const float e = expf(img[i] - M);
        img[i] = e;
        sum += e;
    }
    redv[tid] = sum; __syncthreads();
    for (int s = 128; s > 0; s >>= 1) {
        if (tid < s) redv[tid] += redv[tid + s];
        __syncthreads();
    }
    const float inv = 1.0f / redv[0];
    __syncthreads();

    // normalize; emit prob
    float* prob = out + 5 * B_ + (size_t)b * NS;
    for (int i = tid; i < NS; i += 256) {
        const float p = img[i] * inv;
        img[i]  = p;
        prob[i] = p;
    }
    __syncthreads();

    // integral image: row cumsum then column cumsum
    if (tid < HS) {
        float a = 0.f;
        float* r = img + tid * WS;
        for (int x = 0; x < WS; ++x) { a += r[x]; r[x] = a; }
    }
    __syncthreads();
    if (tid < WS) {
        float a = 0.f;
        for (int y = 0; y < HS; ++y) { a += img[y * WS + tid]; img[y * WS + tid] = a; }
    }
    __syncthreads();

    // 14x14 box sums over 83x83 positions; first-occurrence argmax
    float best = -INFINITY;
    int   bidx = 0x7fffffff;
    for (int i = tid; i < SM * SM; i += 256) {
        const int y = i / SM, x = i % SM;
        const float Dv = img[(y + HT - 1) * WS + (x + WT - 1)];
        const float Bv = (y > 0)           ? img[(y - 1) * WS + (x + WT - 1)] : 0.0f;
        const float Cv = (x > 0)           ? img[(y + HT - 1) * WS + (x - 1)] : 0.0f;
        const float Av = (y > 0 && x > 0)  ? img[(y - 1) * WS + (x - 1)]      : 0.0f;
        const float v = Dv - Bv - Cv + Av;
        if (v > best || (v == best && i < bidx)) { best = v; bidx = i; }
    }
    redv[tid] = best; redi[tid] = bidx; __syncthreads();
    for (int s = 128; s > 0; s >>= 1) {
        if (tid < s) {
            const float v2 = redv[tid + s];
            const int   i2 = redi[tid + s];
            if (v2 > redv[tid] || (v2 == redv[tid] && i2 < redi[tid])) {
                redv[tid] = v2; redi[tid] = i2;
            }
        }
        __syncthreads();
    }
    if (tid == 0) {
        const int idx = redi[0];
        out[0 * B_ + b] = (float)(idx % SM);  // best_x
        out[1 * B_ + b] = (float)(idx / SM);  // best_y
        out[2 * B_ + b] = (float)WT;          // win_w
        out[3 * B_ + b] = (float)HT;          // win_h
        out[4 * B_ + b] = redv[0];            // confidence
    }
}

// ---------------------------------------------------------------------------
extern "C" void kernel_launch(void* const* d_in, const int* in_sizes, int n_in,
                              void* d_out, int out_size, void* d_ws, size_t ws_size,
                              hipStream_t stream) {
    const float* patches = (const float*)d_in[0];   // (2,32,196,768)
    const float* annos   = (const float*)d_in[1];   // (2,32,4)
    const float* skey    = (const float*)d_in[2];   // (32,12,9217,64)
    float*       out     = (float*)d_out;

    // workspace layout (floats): proto | track | blurV  (total 2.4 MB)
    float* ws    = (float*)d_ws;
    float* proto = ws;                       // B_*D_   = 24576
    float* track = proto + (size_t)B_ * D_;  // B_*NS   = 294912
    float* blurV = track + (size_t)B_ * NS;  // B_*NS   = 294912

    proto_kernel<<<B_, 256, 0, stream>>>(patches, annos, proto);
    attn_wmma_kernel<<<dim3(NS / (16 * 8), B_), 256, 0, stream>>>(skey, proto, track);
    blur_v_kernel<<<(B_ * NS) / 256, 256, 0, stream>>>(track, blurV);
    finalize_kernel<<<B_, 256, 0, stream>>>(blurV, out);
}